// SpatialSpectralGraph_20031727468900
// MI455X (gfx1250) — compile-verified
//
#include <hip/hip_runtime.h>
#include <math.h>

// Problem constants (from reference)
#define N_CH 7
#define DM   128
#define NP   512
#define NB   64
#define LN_EPS 1e-5f

typedef __attribute__((ext_vector_type(2))) float v2f;
typedef __attribute__((ext_vector_type(8))) float v8f;

// ---------------------------------------------------------------------------
// Kernel 1: mean pool over patches.  x:(B,C,N,d) -> hpool:(B,C,d)
// One block per (b,c); 1024 threads = 32 patch-slices x 32 float4 lanes.
// Fully coalesced float4 reads; LDS tree reduction over slices.
// ---------------------------------------------------------------------------
__global__ __launch_bounds__(1024) void pool_kernel(const float* __restrict__ x,
                                                    float* __restrict__ hpool) {
  const int bc    = blockIdx.x;        // 0..447
  const int tid   = threadIdx.x;
  const int d4    = tid & 31;          // float4 column within 128-d row
  const int slice = tid >> 5;          // 0..31 patch slice

  const float4* xrow = (const float4*)x + (size_t)bc * (NP * (DM / 4));
  float4 acc = make_float4(0.f, 0.f, 0.f, 0.f);
  for (int n = slice; n < NP; n += 32) {
    float4 v = xrow[n * (DM / 4) + d4];
    acc.x += v.x; acc.y += v.y; acc.z += v.z; acc.w += v.w;
  }

  __shared__ float4 red[32][32];
  red[slice][d4] = acc;
  __syncthreads();
  for (int s = 16; s >= 1; s >>= 1) {
    if (slice < s) {
      float4 o = red[slice + s][d4];
      float4 m = red[slice][d4];
      m.x += o.x; m.y += o.y; m.z += o.z; m.w += o.w;
      red[slice][d4] = m;
    }
    __syncthreads();
  }
  if (slice == 0) {
    const float inv = 1.0f / (float)NP;
    float4 m = red[0][d4];
    float4* out = (float4*)hpool + (size_t)bc * (DM / 4);
    out[d4] = make_float4(m.x * inv, m.y * inv, m.z * inv, m.w * inv);
  }
}

// ---------------------------------------------------------------------------
// Kernel 2: 2-layer GNN on pooled features.  One block per batch, 8 wave32.
// Linear layer uses V_WMMA_F32_16X16X4_F32 (fp32-precise matrix path).
// ---------------------------------------------------------------------------
__global__ __launch_bounds__(256) void gnn_kernel(const float* __restrict__ hpool,
                                                  const float* __restrict__ adj_norm,
                                                  const float* __restrict__ adj_learn,
                                                  const float* __restrict__ W,     // (2,128,128) (out,in)
                                                  const float* __restrict__ bias,  // (2,128)
                                                  const float* __restrict__ gamma, // (2,128)
                                                  const float* __restrict__ beta,  // (2,128)
                                                  float* __restrict__ hout) {      // (64,7,128)
  const int b    = blockIdx.x;
  const int tid  = threadIdx.x;
  const int lane = tid & 31;
  const int wave = tid >> 5;

  __shared__ float sh_h[16][DM];    // current h, rows 7..15 zero-padded
  __shared__ float sh_hm[16][DM];   // after graph propagation (WMMA A operand)
  __shared__ float sh_new[16][DM];  // GELU(linear) output
  __shared__ float sh_adj[N_CH][N_CH];

  for (int i = tid; i < 16 * DM; i += 256) {
    const int r = i >> 7, c = i & 127;
    sh_h[r][c] = (r < N_CH) ? hpool[((size_t)b * N_CH + r) * DM + c] : 0.0f;
  }
  if (tid < N_CH * N_CH) {
    const float a = adj_learn[tid];
    sh_adj[tid / N_CH][tid % N_CH] = adj_norm[tid] + 1.0f / (1.0f + expf(-a));
  }
  __syncthreads();

  for (int l = 0; l < 2; ++l) {
    // ---- graph propagation: hm = adj @ h  (7x7 * 7x128, VALU is right-sized)
    for (int i = tid; i < 16 * DM; i += 256) {
      const int r = i >> 7, c = i & 127;
      float acc = 0.0f;
      if (r < N_CH) {
#pragma unroll
        for (int j = 0; j < N_CH; ++j) acc += sh_adj[r][j] * sh_h[j][c];
      }
      sh_hm[r][c] = acc;
    }
    __syncthreads();

    // ---- linear: lin = hm @ W^T via v_wmma_f32_16x16x4_f32
    // wave w owns output columns [16w, 16w+16); K=128 chained in 32 WMMAs.
    const float* Wl   = W + (size_t)l * DM * DM;
    const int    cb   = wave * 16;
    const int    half = lane >> 4;       // 0: lanes 0-15, 1: lanes 16-31
    const int    klo  = half * 2;        // A/B K sub-offset per ISA layout
    const int    mn   = lane & 15;       // A row / B col within tile
    const float* wrow = Wl + (size_t)(cb + mn) * DM;  // W row for output col cb+mn

    v8f acc = {};
#pragma unroll
    for (int k0 = 0; k0 < DM; k0 += 4) {
      v2f av, bv;
      av.x = sh_hm[mn][k0 + klo];
      av.y = sh_hm[mn][k0 + klo + 1];
      bv.x = wrow[k0 + klo];
      bv.y = wrow[k0 + klo + 1];
      acc = __builtin_amdgcn_wmma_f32_16x16x4_f32(
          /*neg_a=*/false, av, /*neg_b=*/false, bv,
          /*c_mod=*/(short)0, acc, /*reuse_a=*/false, /*reuse_b=*/false);
    }

    // ---- bias + exact GELU (erf form, matches torch F.gelu)
    {
      const int   col = cb + mn;
      const float bv  = bias[l * DM + col];
#pragma unroll
      for (int v = 0; v < 8; ++v) {
        const int   row = v + half * 8;  // C/D layout: VGPR v -> M = v (+8 for hi half)
        const float t   = acc[v] + bv;
        sh_new[row][col] = 0.5f * t * (1.0f + erff(t * 0.70710678118654752f));
      }
    }
    __syncthreads();

    // ---- residual + layernorm: one wave per channel row (rows 0..6)
    if (wave < N_CH) {
      const int r  = wave;
      const int c0 = lane * 4;
      float v0 = sh_h[r][c0 + 0] + sh_new[r][c0 + 0];
      float v1 = sh_h[r][c0 + 1] + sh_new[r][c0 + 1];
      float v2 = sh_h[r][c0 + 2] + sh_new[r][c0 + 2];
      float v3 = sh_h[r][c0 + 3] + sh_new[r][c0 + 3];
      float s = v0 + v1 + v2 + v3;
      for (int off = 16; off >= 1; off >>= 1) s += __shfl_xor(s, off, 32);
      const float mu = s * (1.0f / DM);
      float q = (v0 - mu) * (v0 - mu) + (v1 - mu) * (v1 - mu) +
                (v2 - mu) * (v2 - mu) + (v3 - mu) * (v3 - mu);
      for (int off = 16; off >= 1; off >>= 1) q += __shfl_xor(q, off, 32);
      const float rn = rsqrtf(q * (1.0f / DM) + LN_EPS);
      const float* ga = gamma + l * DM;
      const float* be = beta + l * DM;
      sh_h[r][c0 + 0] = (v0 - mu) * rn * ga[c0 + 0] + be[c0 + 0];
      sh_h[r][c0 + 1] = (v1 - mu) * rn * ga[c0 + 1] + be[c0 + 1];
      sh_h[r][c0 + 2] = (v2 - mu) * rn * ga[c0 + 2] + be[c0 + 2];
      sh_h[r][c0 + 3] = (v3 - mu) * rn * ga[c0 + 3] + be[c0 + 3];
    }
    __syncthreads();
  }

  for (int i = tid; i < N_CH * DM; i += 256)
    hout[(size_t)b * N_CH * DM + i] = sh_h[i >> 7][i & 127];
}

// ---------------------------------------------------------------------------
// Kernel 3: out[b,c,n,:] = x[b,c,n,:] + h[b,c,:]   (bandwidth-dominant pass)
// float4 vectorized; x should still be L2-resident from the pool pass.
// ---------------------------------------------------------------------------
__global__ __launch_bounds__(256) void add_bcast_kernel(const float* __restrict__ x,
                                                        const float* __restrict__ hfin,
                                                        float* __restrict__ out) {
  const size_t idx = (size_t)blockIdx.x * blockDim.x + threadIdx.x;  // float4 index
  const float4* x4 = (const float4*)x;
  const float4* h4 = (const float4*)hfin;
  float4* o4 = (float4*)out;

  const int    d4 = (int)(idx & 31);   // float4 column in 128-d row
  const size_t bc = idx >> 14;         // (idx / 32) / 512  -> (b*7 + c)

  const float4 xv = x4[idx];
  const float4 hv = h4[bc * 32 + d4];
  o4[idx] = make_float4(xv.x + hv.x, xv.y + hv.y, xv.z + hv.z, xv.w + hv.w);
}

// ---------------------------------------------------------------------------
extern "C" void kernel_launch(void* const* d_in, const int* in_sizes, int n_in,
                              void* d_out, int out_size, void* d_ws, size_t ws_size,
                              hipStream_t stream) {
  const float* x         = (const float*)d_in[0];  // (64,7,512,128)
  const float* adj_norm  = (const float*)d_in[1];  // (7,7)
  const float* adj_learn = (const float*)d_in[2];  // (7,7)
  const float* W         = (const float*)d_in[3];  // (2,128,128)
  const float* bias      = (const float*)d_in[4];  // (2,128)
  const float* gamma     = (const float*)d_in[5];  // (2,128)
  const float* beta      = (const float*)d_in[6];  // (2,128)
  float* out = (float*)d_out;

  float* hpool = (float*)d_ws;                     // (64,7,128)
  float* hfin  = hpool + (size_t)NB * N_CH * DM;   // (64,7,128)

  pool_kernel<<<NB * N_CH, 1024, 0, stream>>>(x, hpool);
  gnn_kernel<<<NB, 256, 0, stream>>>(hpool, adj_norm, adj_learn, W, bias, gamma, beta, hfin);

  const size_t n4 = (size_t)NB * N_CH * NP * (DM / 4);  // 7,340,032 float4s
  add_bcast_kernel<<<(unsigned)(n4 / 256), 256, 0, stream>>>(x, hfin, out);
}